// ChannelDirichletNLL_31696858644984
// MI455X (gfx1250) — compile-verified
//
#include <hip/hip_runtime.h>

// ChannelDirichletNLL for B=64, C=4, H=W=256 (N=65536 per channel).
// Memory-bound streaming reduction: 128 MiB in, 4 B out (~5.8 us @ 23.3 TB/s).
// Wave-level reductions use v_wmma_f32_16x16x4_f32 with a ones B-matrix
// (matmul-as-reduction), the CDNA5-native path.

typedef float v2f __attribute__((ext_vector_type(2)));
typedef float v4f __attribute__((ext_vector_type(4)));
typedef float v8f __attribute__((ext_vector_type(8)));

constexpr int kN           = 256 * 256;              // elements per channel
constexpr int kCh          = 64 * 4;                 // 256 channels
constexpr int kBlocksPerCh = 8;
constexpr int kBlocks      = kCh * kBlocksPerCh;     // 2048 blocks
constexpr int kThreads     = 256;                    // 8 waves (wave32)
constexpr int kV4PerBlock  = (kN / kBlocksPerCh) / 4; // 2048 float4 per block
constexpr int kIters       = kV4PerBlock / kThreads;  // 8 float4 per thread

__device__ __forceinline__ float fast_logf(float x) {
  // v_log_f32 computes log2; scale to natural log.
  return 0.69314718055994530942f * __builtin_amdgcn_logf(x);
}

// lgamma(x) for x > 0, branch-free: shift by 6 then Stirling with 1/z corrections.
// lgamma(x) = stirling(x+6) - log(x(x+1)(x+2)(x+3)(x+4)(x+5))
__device__ __forceinline__ float lgammaf_pos(float x) {
  float p = x * (x + 1.f) * (x + 2.f) * (x + 3.f) * (x + 4.f) * (x + 5.f);
  float z = x + 6.f;
  float lz   = fast_logf(z);
  float invz = __builtin_amdgcn_rcpf(z);
  float iz2  = invz * invz;
  // 1/(12z) - 1/(360 z^3) + 1/(1260 z^5)
  float corr = invz * (8.33333333e-2f - iz2 * (2.77777778e-3f - iz2 * 7.93650794e-4f));
  // 0.5*ln(2*pi) = 0.9189385332...
  return (z - 0.5f) * lz - z + 0.91893853320467274f + corr - fast_logf(p);
}

// Full-wave (32 lane) sum via WMMA: D = A x Ones, A[m][k] carries lane values
// in k=0 (VGPR0), k=1/3 zeroed. Row m of D = v[m] + v[m+16]; summing the 8
// D VGPRs gives half-sums in lanes 0-15 vs 16-31; one shfl_xor(16) completes.
__device__ __forceinline__ float wave_sum_wmma(float v) {
  v2f a; a[0] = v;   a[1] = 0.0f;
  v2f b; b[0] = 1.0f; b[1] = 1.0f;   // all-ones B: layout-independent
  v8f c = {};
  v8f d = __builtin_amdgcn_wmma_f32_16x16x4_f32(
      /*neg_a=*/false, a, /*neg_b=*/false, b,
      /*c_mod=*/(short)0, c, /*reuse_a=*/false, /*reuse_b=*/false);
  float s = ((d[0] + d[1]) + (d[2] + d[3])) + ((d[4] + d[5]) + (d[6] + d[7]));
  s += __shfl_xor(s, 16, 32);
  return s;
}

__global__ __launch_bounds__(kThreads)
void dirichlet_partial_kernel(const float* __restrict__ x_hat,
                              const float* __restrict__ x,
                              float* __restrict__ t_part,
                              float* __restrict__ a_part) {
  const int  blk  = blockIdx.x;
  const int  ch   = blk / kBlocksPerCh;
  const int  sub  = blk % kBlocksPerCh;
  const long base4 = (long)ch * (kN / 4) + (long)sub * kV4PerBlock;

  const v4f* a4 = (const v4f*)x_hat + base4;
  const v4f* x4 = (const v4f*)x + base4;

  float t_acc = 0.f;
  float a_acc = 0.f;

#pragma unroll
  for (int k = 0; k < kIters; ++k) {
    const int i = k * kThreads + threadIdx.x;          // coalesced b128 loads
    v4f av = __builtin_nontemporal_load(&a4[i]);       // streaming: NT hint
    v4f xv = __builtin_nontemporal_load(&x4[i]);
    t_acc += (av[0] - 1.f) * fast_logf(xv[0] + 1e-8f) - lgammaf_pos(av[0]);
    t_acc += (av[1] - 1.f) * fast_logf(xv[1] + 1e-8f) - lgammaf_pos(av[1]);
    t_acc += (av[2] - 1.f) * fast_logf(xv[2] + 1e-8f) - lgammaf_pos(av[2]);
    t_acc += (av[3] - 1.f) * fast_logf(xv[3] + 1e-8f) - lgammaf_pos(av[3]);
    a_acc += (av[0] + av[1]) + (av[2] + av[3]);
  }

  const float wt = wave_sum_wmma(t_acc);
  const float wa = wave_sum_wmma(a_acc);

  __shared__ float sh_t[kThreads / 32];
  __shared__ float sh_a[kThreads / 32];
  const int wave = threadIdx.x >> 5;
  const int lane = threadIdx.x & 31;
  if (lane == 0) { sh_t[wave] = wt; sh_a[wave] = wa; }
  __syncthreads();
  if (threadIdx.x == 0) {
    float T = 0.f, A = 0.f;
#pragma unroll
    for (int i = 0; i < kThreads / 32; ++i) { T += sh_t[i]; A += sh_a[i]; }
    t_part[blk] = T;
    a_part[blk] = A;
  }
}

__global__ __launch_bounds__(256)
void dirichlet_final_kernel(const float* __restrict__ t_part,
                            const float* __restrict__ a_part,
                            float* __restrict__ out) {
  const int c = threadIdx.x;  // one thread per (b,c) channel, 256 total
  float ts = 0.f;
  float As = 0.f;
#pragma unroll
  for (int j = 0; j < kBlocksPerCh; ++j) {
    ts += t_part[c * kBlocksPerCh + j];
    As += a_part[c * kBlocksPerCh + j];
  }
  // lgamma(sum_a), sum_a ~ 65536: double-precision Stirling (256 evals only,
  // absolute magnitude ~6.6e5 so fp32 log would lose too much here).
  const double Ad = (double)As;
  const double lg = (Ad - 0.5) * log(Ad) - Ad
                  + 0.91893853320467274178 + 1.0 / (12.0 * Ad);
  const float v = ts + (float)lg;   // = log_prob for this channel

  const float w = wave_sum_wmma(v);
  __shared__ float sh[8];
  const int wave = threadIdx.x >> 5;
  const int lane = threadIdx.x & 31;
  if (lane == 0) sh[wave] = w;
  __syncthreads();
  if (threadIdx.x == 0) {
    float S = 0.f;
#pragma unroll
    for (int i = 0; i < 8; ++i) S += sh[i];
    out[0] = -S / 256.0f;   // mean of negative log_prob
  }
}

extern "C" void kernel_launch(void* const* d_in, const int* in_sizes, int n_in,
                              void* d_out, int out_size, void* d_ws, size_t ws_size,
                              hipStream_t stream) {
  const float* x_hat = (const float*)d_in[0];  // concentrations a
  const float* x     = (const float*)d_in[1];  // simplex samples
  float* ws     = (float*)d_ws;
  float* t_part = ws;               // [kBlocks]
  float* a_part = ws + kBlocks;     // [kBlocks]  (16 KB total)

  dirichlet_partial_kernel<<<kBlocks, kThreads, 0, stream>>>(x_hat, x, t_part, a_part);
  dirichlet_final_kernel<<<1, 256, 0, stream>>>(t_part, a_part, (float*)d_out);
}